// TopologyBatch_76699525972632
// MI455X (gfx1250) — compile-verified
//
#include <hip/hip_runtime.h>
#include <math.h>

#define FR      64      // frames staged per block
#define JSTRIDE 127     // dwords per frame slot in LDS (126 data + 1 pad; 127 coprime with 64 banks)
#define GROUPS  4       // path-pair worker groups per block
#define OUTPF   63      // velocity outputs per block (needs FR frames)
#define BATCH   32
#define FRAMES  1024
#define CHUNKS  (((FRAMES - 1) + OUTPF - 1) / OUTPF)   // 17

// PATHS = chain[1:] for each kinematic chain
__constant__ int cPATH[5][6] = {
    { 2,  5,  8, 11,  0, 0},
    { 1,  4,  7, 10,  0, 0},
    { 3,  6,  9, 12, 15, 0},
    {14, 17, 19, 21,  0, 0},
    {13, 16, 18, 20,  0, 0}};
__constant__ int cLEN[5] = {4, 4, 5, 4, 4};

struct V3 { float x, y, z; };

static __device__ __forceinline__ V3 v3(float x, float y, float z) { V3 r; r.x = x; r.y = y; r.z = z; return r; }
static __device__ __forceinline__ V3 subv(V3 a, V3 b) { return v3(a.x - b.x, a.y - b.y, a.z - b.z); }
static __device__ __forceinline__ float dotv(V3 a, V3 b) { return fmaf(a.x, b.x, fmaf(a.y, b.y, a.z * b.z)); }
static __device__ __forceinline__ V3 crossv(V3 a, V3 b) {
    return v3(fmaf(a.y, b.z, -a.z * b.y),
              fmaf(a.z, b.x, -a.x * b.z),
              fmaf(a.x, b.y, -a.y * b.x));
}
static __device__ __forceinline__ V3 nrm(V3 f) {
    float l2 = dotv(f, f);
    float s  = (l2 > 0.0f) ? __builtin_amdgcn_rsqf(l2) : 0.0f;   // v_rsq_f32
    return v3(f.x * s, f.y * s, f.z * s);
}
static __device__ __forceinline__ float clampd(float x) {
    const float lo = -1.0f + 1e-7f, hi = 1.0f - 1e-7f;
    return fminf(fmaxf(x, lo), hi);
}
static __device__ __forceinline__ V3 ldp(const float* p) { return v3(p[0], p[1], p[2]); }

// Gauss linking integral for one segment pair; returns gli/(4*pi) with sign applied.
static __device__ __forceinline__ float gauss(V3 s1, V3 e1, V3 s2, V3 e2) {
    V3 r13 = subv(s2, s1);
    V3 r14 = subv(e2, s1);
    V3 r23 = subv(s2, e1);
    V3 r24 = subv(e2, e1);
    V3 r12 = subv(e1, s1);
    V3 r34 = subv(e2, s2);
    V3 n0 = nrm(crossv(r13, r14));
    V3 n1 = nrm(crossv(r14, r24));
    V3 n2 = nrm(crossv(r24, r23));
    V3 n3 = nrm(crossv(r23, r13));
    float s = asinf(clampd(dotv(n0, n1))) + asinf(clampd(dotv(n1, n2))) +
              asinf(clampd(dotv(n2, n3))) + asinf(clampd(dotv(n3, n0)));
    float sg = dotv(crossv(r34, r12), r13);
    s = (sg <= 0.0f) ? -s : s;
    return s * 0.07957747154594767f;   // 1/(4*pi)
}

__global__ __launch_bounds__(FR * GROUPS)
void gli_vel_kernel(const float* __restrict__ m1, const float* __restrict__ m2,
                    float* __restrict__ out) {
    __shared__ float stage[FR * JSTRIDE];   // 32512 B: per-frame joints (m1[1..21], m2[1..21])
    __shared__ float gsm[FR * 25];          //  6400 B: per-frame gli[25]

    const int tid  = threadIdx.x;
    const int t    = tid & (FR - 1);   // frame lane within block
    const int g    = tid >> 6;         // path-pair worker group 0..3 (wave-uniform)
    const int b    = blockIdx.y;
    const int base = blockIdx.x * OUTPF;
    int f = base + t;
    if (f > FRAMES - 1) f = FRAMES - 1;   // clamp tail; keeps control flow uniform

    // ---- stage 126 dwords/frame into LDS via gfx1250 async global->LDS loads ----
    // layout per frame slot: dword d in [0,63) = motion1 joint(1+d/3) comp(d%3),
    //                        dword d in [63,126) = motion2 likewise.
    const size_t frameOff = ((size_t)b * FRAMES + (size_t)f) * 66 + 3;  // skip joint 0
    const float* src1 = m1 + frameOff;
    const float* src2 = m2 + frameOff;
    const unsigned ldsBase = (unsigned)(size_t)(&stage[t * JSTRIDE]);   // low 32 bits = LDS offset

    for (int k = 0; k < 32; ++k) {
        int d = g * 32 + k;                       // wave-uniform
        if (d < 126) {
            const float* src = (d < 63) ? (src1 + d) : (src2 + (d - 63));
            unsigned lv = ldsBase + (unsigned)d * 4u;
            unsigned long long ga = (unsigned long long)(size_t)src;
            asm volatile("global_load_async_to_lds_b32 %0, %1, off"
                         :: "v"(lv), "v"(ga) : "memory");
        }
    }
    asm volatile("s_wait_asynccnt 0x0" ::: "memory");
    __syncthreads();

    const float* P1 = &stage[t * JSTRIDE];
    const float* P2 = P1 + 63;

    // ---- each group computes path-pairs q = g, g+4, ... (25 total) ----
    for (int q = g; q < 25; q += GROUPS) {
        int a = q / 5;
        int c = q - a * 5;
        int na = cLEN[a] - 1;
        int nc = cLEN[c] - 1;
        float acc = 0.0f;
        for (int i = 0; i < na; ++i) {
            V3 s1 = ldp(P1 + (cPATH[a][i]     - 1) * 3);
            V3 e1 = ldp(P1 + (cPATH[a][i + 1] - 1) * 3);
            for (int j = 0; j < nc; ++j) {
                V3 s2 = ldp(P2 + (cPATH[c][j]     - 1) * 3);
                V3 e2 = ldp(P2 + (cPATH[c][j + 1] - 1) * 3);
                acc += gauss(s1, e1, s2, e2);
            }
        }
        gsm[t * 25 + q] = acc;
    }
    __syncthreads();

    // ---- temporal velocity + max over 25 path-pairs ----
    if (g == 0 && t < OUTPF) {
        int fo = base + t;
        if (fo < FRAMES - 1) {
            const float* ga_ = &gsm[t * 25];
            const float* gb_ = &gsm[(t + 1) * 25];
            float mx = 0.0f;
            #pragma unroll
            for (int q = 0; q < 25; ++q) mx = fmaxf(mx, fabsf(gb_[q] - ga_[q]));
            out[(size_t)b * (FRAMES - 1) + fo] = mx;
        }
    }
}

extern "C" void kernel_launch(void* const* d_in, const int* in_sizes, int n_in,
                              void* d_out, int out_size, void* d_ws, size_t ws_size,
                              hipStream_t stream) {
    (void)in_sizes; (void)n_in; (void)out_size; (void)d_ws; (void)ws_size;
    const float* motion1 = (const float*)d_in[0];
    const float* motion2 = (const float*)d_in[1];
    float* out = (float*)d_out;
    dim3 grid(CHUNKS, BATCH, 1);
    gli_vel_kernel<<<grid, FR * GROUPS, 0, stream>>>(motion1, motion2, out);
}